// GraphLayer_68719477425
// MI455X (gfx1250) — compile-verified
//
#include <hip/hip_runtime.h>
#include <hip/hip_bf16.h>

// ---------------------------------------------------------------------------
// GraphLayer for MI455X (gfx1250, wave32).
//   B=8, N=256, D=128, BN=2048.
//   GEMMs -> v_wmma_f32_16x16x4_f32 (fp32 matrix core path). Variants are
//   compile-time templated so the K-loop is branch-free and pipelines
//   (branchy runtime 'if' inside the loop wrecked clauses in round 2).
//   Edge aggregation (relu couples i,j,d -> not a matmul) -> VALU FMA loop,
//   4 receiver nodes per block to cut L2 re-read traffic of HJ by 4x.
// ---------------------------------------------------------------------------

typedef float v2f __attribute__((ext_vector_type(2)));
typedef float v8f __attribute__((ext_vector_type(8)));

#define BN    2048
#define NDIM  128
#define KDIM  128
#define NNODE 256
#define TILEM 16
#define LDA_P 132   // padded LDS row stride (floats) to avoid bank conflicts
#define ITILE 4     // receiver nodes per edge_agg block

// Generic tile GEMM over a 16x128 output tile. Block = 256 threads = 8 waves;
// wave w owns output columns [16w,16w+16).
//   acc1 = A@W   (+ A2@W2 if HAS_A2)  -> C  with (bias*rowscale, addsrc, relu)
//   acc2 = A@Wb  (if DUAL)            -> Cb (raw, shares A fragments)
template <bool DUAL, bool HAS_A2>
__global__ __launch_bounds__(256) void gemm16_wmma(
    const float* __restrict__ A,   const float* __restrict__ W,
    const float* __restrict__ Wb,        // second weight for dual output
    const float* __restrict__ A2,  const float* __restrict__ W2,   // summed pair
    const float* __restrict__ bias,      // [128] or null
    const float* __restrict__ rowscale,  // [BN] scales bias per row, or null
    const float* __restrict__ addsrc,    // [BN,128] residual, or null
    float* __restrict__ C, float* __restrict__ Cb,
    int relu_flag)
{
    __shared__ float sA [TILEM * LDA_P];
    __shared__ float sA2[TILEM * LDA_P];

    const int tid   = threadIdx.x;
    const int lane  = tid & 31;
    const int wave  = tid >> 5;
    const int mbase = blockIdx.x * TILEM;

    // Stage A tile (16x128 f32) into LDS: 256 threads x 8 floats, float4 x2.
    {
        const int r  = tid >> 4;          // 0..15
        const int c0 = (tid & 15) * 8;    // 0..120
        const float4* src = reinterpret_cast<const float4*>(
            A + (size_t)(mbase + r) * KDIM + c0);
        float4 x0 = src[0], x1 = src[1];
        float4* dst = reinterpret_cast<float4*>(&sA[r * LDA_P + c0]);
        dst[0] = x0; dst[1] = x1;
        if (HAS_A2) {
            const float4* src2 = reinterpret_cast<const float4*>(
                A2 + (size_t)(mbase + r) * KDIM + c0);
            float4 y0 = src2[0], y1 = src2[1];
            float4* dst2 = reinterpret_cast<float4*>(&sA2[r * LDA_P + c0]);
            dst2[0] = y0; dst2[1] = y1;
        }
    }
    __syncthreads();

    // WMMA fragment coordinates (ISA 7.12.2, 32-bit 16x16x4):
    //   A: lanes 0-15 -> rows, v0/v1 = K 0/1; lanes 16-31 -> K 2/3.
    //   B: mirrored (lane = column, K split at lane 16).
    //   C/D: VGPR r -> row r (lanes<16) / r+8 (lanes>=16), col = lane&15.
    const int kb  = (lane >> 4) << 1;           // 0 or 2
    const int row = lane & 15;
    const int col = (wave << 4) + (lane & 15);  // 0..127

    v8f acc1 = {};
    v8f acc2 = {};
    #pragma unroll 8
    for (int kk = 0; kk < KDIM; kk += 4) {
        v2f a, b;
        a.x = sA[row * LDA_P + kk + kb];
        a.y = sA[row * LDA_P + kk + kb + 1];
        b.x = W[(size_t)(kk + kb)     * NDIM + col];
        b.y = W[(size_t)(kk + kb + 1) * NDIM + col];
        acc1 = __builtin_amdgcn_wmma_f32_16x16x4_f32(
            false, a, false, b, (short)0, acc1, false, false);
        if (DUAL) {     // compile-time: reuse the same A fragment
            v2f b2;
            b2.x = Wb[(size_t)(kk + kb)     * NDIM + col];
            b2.y = Wb[(size_t)(kk + kb + 1) * NDIM + col];
            acc2 = __builtin_amdgcn_wmma_f32_16x16x4_f32(
                false, a, false, b2, (short)0, acc2, false, false);
        }
    }
    if (HAS_A2) {
        #pragma unroll 8
        for (int kk = 0; kk < KDIM; kk += 4) {
            v2f a, b;
            a.x = sA2[row * LDA_P + kk + kb];
            a.y = sA2[row * LDA_P + kk + kb + 1];
            b.x = W2[(size_t)(kk + kb)     * NDIM + col];
            b.y = W2[(size_t)(kk + kb + 1) * NDIM + col];
            acc1 = __builtin_amdgcn_wmma_f32_16x16x4_f32(
                false, a, false, b, (short)0, acc1, false, false);
        }
    }

    // Epilogue (uniform runtime branches; outside the hot loop)
    #pragma unroll
    for (int r = 0; r < 8; ++r) {
        const int orow = (lane < 16) ? r : (r + 8);
        const int grow = mbase + orow;
        float v = acc1[r];
        if (bias) {
            const float bs = rowscale ? rowscale[grow] : 1.0f;
            v += bias[col] * bs;
        }
        if (addsrc) v += addsrc[(size_t)grow * NDIM + col];
        if (relu_flag) v = fmaxf(v, 0.0f);
        C[(size_t)grow * NDIM + col] = v;
        if (DUAL) Cb[(size_t)grow * NDIM + col] = acc2[r];
    }
}

// Edge-message aggregation (the N^2*D part — inherently VALU, not a matmul):
//   S[b,i,d]   = sum_j a'[b,i,j] * relu(HI[b,i,d] + HJ[b,j,d])  (a' zeroes diag)
//   rowsum[bi] = sum_j a'[b,i,j]
// One block per (b, 4 consecutive i); thread d = 0..127. Each coalesced HJ
// row load is reused for 4 receivers (4x less L2 traffic); adjacency reads
// are block-uniform -> scalar cache path.
__global__ __launch_bounds__(128) void edge_agg(
    const float* __restrict__ HI,    // [BN,128], b1 pre-folded
    const float* __restrict__ HJ,    // [BN,128]
    const float* __restrict__ Adj,   // [B,N,N]
    float* __restrict__ S,           // [BN,128]
    float* __restrict__ rowsum)      // [BN]
{
    const int bi0 = blockIdx.x * ITILE;   // first receiver row, 0..2044
    const int b   = bi0 >> 8;
    const int i0  = bi0 & (NNODE - 1);
    const int d   = threadIdx.x;

    const float* __restrict__ hjb   = HJ  + (size_t)b * NNODE * NDIM;
    const float* __restrict__ abase = Adj + ((size_t)b * NNODE + i0) * NNODE;

    float hi[ITILE], s[ITILE], rs[ITILE];
    #pragma unroll
    for (int t = 0; t < ITILE; ++t) {
        hi[t] = HI[(size_t)(bi0 + t) * NDIM + d];
        s[t]  = 0.0f;
        rs[t] = 0.0f;
    }

    #pragma unroll 2
    for (int j = 0; j < NNODE; ++j) {
        const float hj = hjb[(size_t)j * NDIM + d];
        #pragma unroll
        for (int t = 0; t < ITILE; ++t) {
            const float a = (j == i0 + t) ? 0.0f : abase[(size_t)t * NNODE + j];
            const float h = hi[t] + hj;
            s[t]  = fmaf(a, fmaxf(h, 0.0f), s[t]);
            rs[t] += a;
        }
    }

    #pragma unroll
    for (int t = 0; t < ITILE; ++t) {
        S[(size_t)(bi0 + t) * NDIM + d] = s[t];
        if (d == 0) rowsum[bi0 + t] = rs[t];
    }
}

// Row LayerNorm over D=128; one block (4 waves) per row.
__global__ __launch_bounds__(128) void layernorm_row(
    const float* __restrict__ RES,
    const float* __restrict__ g, const float* __restrict__ bt,
    float* __restrict__ out)
{
    const int rowid = blockIdx.x;
    const int d     = threadIdx.x;
    const float x = RES[(size_t)rowid * NDIM + d];

    float s = x, s2 = x * x;
    #pragma unroll
    for (int off = 16; off > 0; off >>= 1) {
        s  += __shfl_down(s,  off);
        s2 += __shfl_down(s2, off);
    }
    __shared__ float red[2][4];
    const int lane = d & 31, w = d >> 5;
    if (lane == 0) { red[0][w] = s; red[1][w] = s2; }
    __syncthreads();
    const float ts  = red[0][0] + red[0][1] + red[0][2] + red[0][3];
    const float ts2 = red[1][0] + red[1][1] + red[1][2] + red[1][3];
    const float mu  = ts * (1.0f / NDIM);
    const float var = ts2 * (1.0f / NDIM) - mu * mu;
    const float inv = rsqrtf(var + 1e-5f);
    out[(size_t)rowid * NDIM + d] = (x - mu) * inv * g[d] + bt[d];
}

extern "C" void kernel_launch(void* const* d_in, const int* in_sizes, int n_in,
                              void* d_out, int out_size, void* d_ws, size_t ws_size,
                              hipStream_t stream) {
    const float* X    = (const float*)d_in[0];               // [8,256,128]
    const float* Adj  = (const float*)d_in[1];               // [8,256,256]
    const float* W1   = (const float*)d_in[2];               // [256,128]
    const float* b1   = (const float*)d_in[3];               // [128]
    const float* W2   = (const float*)d_in[4];               // [128,128]
    const float* b2   = (const float*)d_in[5];               // [128]
    const float* U1   = (const float*)d_in[6];               // [256,128]
    const float* ub1  = (const float*)d_in[7];               // [128]
    const float* U2   = (const float*)d_in[8];               // [128,128]
    const float* ub2  = (const float*)d_in[9];               // [128]
    const float* ln_g = (const float*)d_in[10];              // [128]
    const float* ln_b = (const float*)d_in[11];              // [128]
    float* out = (float*)d_out;

    const float* W1a = W1;                 // msg_w1[:128]
    const float* W1b = W1 + 128 * 128;     // msg_w1[128:]
    const float* U1a = U1;                 // upd_w1[:128]
    const float* U1b = U1 + 128 * 128;     // upd_w1[128:]

    // Workspace: three reusable [2048,128] f32 buffers + rowsum. ~3.2 MB.
    const size_t MAT = (size_t)BN * NDIM;
    float* buf0   = (float*)d_ws;          // HI  -> AGG
    float* buf1   = buf0 + MAT;            // HJ  -> H1
    float* buf2   = buf1 + MAT;            // S   -> RES
    float* rowsum = buf2 + MAT;            // [2048]

    dim3 gblk(256), ggrd(BN / TILEM);      // 128 blocks, 8 waves each

    // 1) HI = X @ W1a + b1   and   HJ = X @ W1b   (dual output, shared A)
    gemm16_wmma<true, false><<<ggrd, gblk, 0, stream>>>(
        X, W1a, W1b, nullptr, nullptr, b1, nullptr, nullptr, buf0, buf1, 0);
    // 2) S = sum_j a'_{ij} relu(HI_i + HJ_j);  rowsum = sum_j a'_{ij}
    edge_agg<<<dim3(BN / ITILE), dim3(128), 0, stream>>>(buf0, buf1, Adj,
                                                         buf2, rowsum);
    // 3) AGG = S @ W2 + b2 * rowsum          (writes over HI slot)
    gemm16_wmma<false, false><<<ggrd, gblk, 0, stream>>>(
        buf2, W2, nullptr, nullptr, nullptr, b2, rowsum, nullptr,
        buf0, nullptr, 0);
    // 4) H1 = relu(X @ U1a + AGG @ U1b + ub1) (writes over HJ slot)
    gemm16_wmma<false, true><<<ggrd, gblk, 0, stream>>>(
        X, U1a, nullptr, buf0, U1b, ub1, nullptr, nullptr, buf1, nullptr, 1);
    // 5) RES = H1 @ U2 + ub2 + X              (writes over S slot)
    gemm16_wmma<false, false><<<ggrd, gblk, 0, stream>>>(
        buf1, U2, nullptr, nullptr, nullptr, ub2, nullptr, X, buf2, nullptr, 0);
    // 6) out = LayerNorm(RES) * g + b
    layernorm_row<<<dim3(BN), dim3(128), 0, stream>>>(buf2, ln_g, ln_b, out);
}